// ConditionalNTXentLoss_6227702579662
// MI455X (gfx1250) — compile-verified
//
#include <hip/hip_runtime.h>
#include <hip/hip_bf16.h>

typedef __attribute__((ext_vector_type(2))) float v2f;
typedef __attribute__((ext_vector_type(8))) float v8f;

#define TEMP_INV 2.0f     // 1 / TEMPERATURE (T = 0.5)
#define COS_EPS  1e-8f
#define D        256      // feature dim
#define WAVES_PER_BLOCK 8

// Each wave handles 4 chunks = 16 rows of length 256.
// Tile R (16x256): row m -> chunk c0+(m>>2), within=m&3;
//   within<2 -> zjs[2*chunk+within], else zis[2*chunk+within-2]
// G = R * R^T computed with 64 accumulated V_WMMA_F32_16X16X4_F32 steps.
// Per-lane operand for both A (16x4) and B (4x16) slices at K-step k:
//   lane L supplies R[L&15][4k + 2*(L>>4)] and R[L&15][4k + 2*(L>>4)+1]
__global__ __launch_bounds__(256) void ntxent_gram_kernel(
    const float* __restrict__ zis, const float* __restrict__ zjs,
    float* __restrict__ partial)
{
    __shared__ float gsm[WAVES_PER_BLOCK][16][16];

    const int lane        = threadIdx.x & 31;
    const int waveInBlock = threadIdx.x >> 5;
    const int tile        = blockIdx.x * WAVES_PER_BLOCK + waveInBlock;
    const int c0          = tile * 4;          // first chunk of this tile

    const int m      = lane & 15;              // tile row this lane feeds
    const int chunk  = c0 + (m >> 2);
    const int within = m & 3;
    const float* rowptr = (within < 2)
        ? (zjs + (size_t)(2 * chunk + within) * D)
        : (zis + (size_t)(2 * chunk + (within - 2)) * D);
    const int colbase = (lane >> 4) * 2;       // K sub-offset for this half-wave

    v8f acc = {0.f, 0.f, 0.f, 0.f, 0.f, 0.f, 0.f, 0.f};
    #pragma unroll 8
    for (int k = 0; k < D / 4; ++k) {
        v2f ab = *(const v2f*)(rowptr + 4 * k + colbase);
        // D = A*B + C ; A == B per-lane data for the symmetric Gram product
        acc = __builtin_amdgcn_wmma_f32_16x16x4_f32(
            /*neg_a=*/false, ab, /*neg_b=*/false, ab,
            /*c_mod=*/(short)0, acc, /*reuse_a=*/false, /*reuse_b=*/false);
    }

    // Spill Gram tile: lane L holds G[v + 8*(L>>4)][L&15] in acc[v]
    const int half = lane >> 4;
    #pragma unroll
    for (int v = 0; v < 8; ++v)
        gsm[waveInBlock][v + 8 * half][m] = acc[v];
    __syncthreads();

    // Lanes 0..15: one loss row each (tile row m).
    float loss = 0.f;
    if (lane < 16) {
        const float (*G)[16] = gsm[waveInBlock];
        const int bb = m & ~3;        // chunk block base row
        const int i  = m & 3;         // row within chunk's 4x4 block
        const int j1 = bb + (i ^ 1);  // negative 1
        const int j2 = bb + (i ^ 2);  // positive
        const int j3 = bb + (i ^ 3);  // negative 2

        const float ni = fmaxf(sqrtf(G[m][m]),   COS_EPS);
        const float n1 = fmaxf(sqrtf(G[j1][j1]), COS_EPS);
        const float n2 = fmaxf(sqrtf(G[j2][j2]), COS_EPS);
        const float n3 = fmaxf(sqrtf(G[j3][j3]), COS_EPS);

        const float p = G[m][j2] / (ni * n2) * TEMP_INV;
        const float a = G[m][j1] / (ni * n1) * TEMP_INV;
        const float b = G[m][j3] / (ni * n3) * TEMP_INV;

        const float mx  = fmaxf(p, fmaxf(a, b));
        const float lse = __logf(__expf(p - mx) + __expf(a - mx) + __expf(b - mx)) + mx;
        loss = lse - p;   // -log_softmax(...)[0]
    }

    // Wave32 reduction (lanes >= 16 contribute 0)
    loss += __shfl_xor(loss, 16);
    loss += __shfl_xor(loss, 8);
    loss += __shfl_xor(loss, 4);
    loss += __shfl_xor(loss, 2);
    loss += __shfl_xor(loss, 1);

    if (lane == 0) partial[tile] = loss;
}

// Deterministic fixed-order final reduction (no float atomics).
__global__ __launch_bounds__(256) void ntxent_reduce_kernel(
    const float* __restrict__ partial, float* __restrict__ out,
    int n, float scale)
{
    __shared__ float red[256];
    float s = 0.f;
    for (int i = threadIdx.x; i < n; i += 256) s += partial[i];
    red[threadIdx.x] = s;
    __syncthreads();
    for (int off = 128; off > 0; off >>= 1) {
        if ((int)threadIdx.x < off) red[threadIdx.x] += red[threadIdx.x + off];
        __syncthreads();
    }
    if (threadIdx.x == 0) out[0] = red[0] * scale;
}

extern "C" void kernel_launch(void* const* d_in, const int* in_sizes, int n_in,
                              void* d_out, int out_size, void* d_ws, size_t ws_size,
                              hipStream_t stream) {
    const float* zis = (const float*)d_in[0];
    const float* zjs = (const float*)d_in[1];
    float* out      = (float*)d_out;
    float* partial  = (float*)d_ws;

    const int B       = in_sizes[0] / D;            // 65536 rows
    const int nChunks = B / 2;                      // 32768
    const int nTiles  = nChunks / 4;                // 8192 (one per wave)
    const int nBlocks = nTiles / WAVES_PER_BLOCK;   // 1024

    ntxent_gram_kernel<<<nBlocks, 256, 0, stream>>>(zis, zjs, partial);
    ntxent_reduce_kernel<<<1, 256, 0, stream>>>(partial, out, nTiles,
                                                1.0f / (float)B);
}